// KnowledgeRetriever_37864431681816
// MI455X (gfx1250) — compile-verified
//
#include <hip/hip_runtime.h>
#include <hip/hip_bf16.h>
#include <math.h>

// ---------------------------------------------------------------------------
// Problem constants (match reference)
// ---------------------------------------------------------------------------
#define H_DIM   1024
#define KDIM    128
#define N_ENT   2048
#define MAX_K   32
#define BS      1024        // B*S = 2*512
#define OUT_EMB_OFF   0
#define OUT_MASK_OFF  (BS * MAX_K * KDIM)               // 4,194,304
#define OUT_SCORE_OFF (OUT_MASK_OFF + BS * MAX_K)       // 4,227,072

typedef __attribute__((ext_vector_type(2))) float v2f;
typedef __attribute__((ext_vector_type(8))) float v8f;

// ---------------------------------------------------------------------------
// Gather entity embeddings: ent[n,k] = emb_table[entity_idx[n], k]
// ---------------------------------------------------------------------------
__global__ void gather_ent_kernel(const float* __restrict__ emb,
                                  const int*   __restrict__ idx,
                                  float*       __restrict__ ent) {
    int i = blockIdx.x * blockDim.x + threadIdx.x;     // n*KDIM + k
    int n = i >> 7;
    int k = i & (KDIM - 1);
    ent[i] = emb[(size_t)idx[n] * KDIM + k];
}

// ---------------------------------------------------------------------------
// bc[j] = sum_k bq[k] * W1a[k,j]   (folded query bias)
// ---------------------------------------------------------------------------
__global__ void bias_fold_kernel(const float* __restrict__ bq,
                                 const float* __restrict__ W1,
                                 float*       __restrict__ bc) {
    int j = threadIdx.x;                               // 128 threads
    float s = 0.f;
    for (int k = 0; k < KDIM; ++k) s += bq[k] * W1[k * KDIM + j];
    bc[j] = s;
}

// ---------------------------------------------------------------------------
// FP32 WMMA GEMM: C[M,N] = A[M,K] @ B[K,N] (+ bias[N]), optional transposed
// store (C_T[N, ldT] with ldT = transLd). One wave per 16x16 output tile.
// Uses V_WMMA_F32_16X16X4_F32 (exact fp32, K stepped by 4).
//
// f32 A 16x4 layout : lane&15 = M row ; (lane>>4)*2 = K pair ; vgpr0/1 = K,K+1
// f32 B 4x16 layout : lane&15 = N col ; (lane>>4)*2 = K pair ; vgpr0/1 = K,K+1
// f32 C 16x16       : lane&15 = N ; vgpr i -> M = i + (lane>>4)*8
// ---------------------------------------------------------------------------
__global__ void wmma_gemm_f32_kernel(const float* __restrict__ A,
                                     const float* __restrict__ B,
                                     const float* __restrict__ bias,
                                     float*       __restrict__ C,
                                     int M, int N, int K, int transLd) {
    const int lane   = threadIdx.x & 31;
    const int wave   = threadIdx.x >> 5;
    const int tileId = blockIdx.x * (blockDim.x >> 5) + wave;
    const int tilesN = N >> 4;
    const int tM     = tileId / tilesN;
    const int tN     = tileId - tM * tilesN;
    if (tM * 16 >= M) return;                          // wave-uniform guard

    const int r  = lane & 15;          // M row (for A) / N col (for B, C)
    const int kh = (lane >> 4) << 1;   // 0 or 2: which K pair this half-wave owns
    const float* __restrict__ Arow = A + (size_t)(tM * 16 + r) * K;
    const float* __restrict__ Bcol = B + (tN * 16 + r);

    v8f c = {};
    for (int k0 = 0; k0 < K; k0 += 4) {
        v2f a, b;
        a.x = Arow[k0 + kh];
        a.y = Arow[k0 + kh + 1];
        b.x = Bcol[(size_t)(k0 + kh) * N];
        b.y = Bcol[(size_t)(k0 + kh + 1) * N];
        // (neg_a, A, neg_b, B, c_mod, C, reuse_a, reuse_b)
        c = __builtin_amdgcn_wmma_f32_16x16x4_f32(false, a, false, b,
                                                  (short)0, c, false, false);
    }

    const float bv = bias ? bias[tN * 16 + r] : 0.f;   // per-column bias
    const int   hi = (lane >> 4) * 8;
    if (transLd == 0) {
        for (int i = 0; i < 8; ++i) {
            int row = tM * 16 + hi + i;
            C[(size_t)row * N + tN * 16 + r] = c[i] + bv;
        }
    } else {
        // store transposed: C_T[col, row], leading dim = transLd
        for (int i = 0; i < 8; ++i) {
            int row = tM * 16 + hi + i;
            C[(size_t)(tN * 16 + r) * transLd + row] = c[i] + bv;
        }
    }
}

// ---------------------------------------------------------------------------
// Scorer + top-32 + gather. One block (256 threads) per (b,s) position.
// scores[n] = sum_k w2[k] * gelu(q[k] + e_t[k][n]) + b2
// ---------------------------------------------------------------------------
__device__ __forceinline__ float gelu_exact(float x) {
    return 0.5f * x * (1.0f + erff(x * 0.70710678118654752f));
}

__global__ void score_topk_kernel(const float* __restrict__ qpart,   // [BS,KDIM]
                                  const float* __restrict__ e_t,     // [KDIM,N_ENT]
                                  const float* __restrict__ ent,     // [N_ENT,KDIM]
                                  const float* __restrict__ W2,      // [KDIM]
                                  const float* __restrict__ b2,      // [1]
                                  float* __restrict__ out) {
    __shared__ float qs[KDIM];
    __shared__ float w2s[KDIM];
    __shared__ float sc[N_ENT];
    __shared__ float rmax[256];
    __shared__ int   rarg[256];
    __shared__ int   sidx[MAX_K];

    const int bs  = blockIdx.x;
    const int tid = threadIdx.x;

    if (tid < KDIM) {
        qs[tid]  = qpart[(size_t)bs * KDIM + tid];
        w2s[tid] = W2[tid];
    }
    __syncthreads();

    const float bias2 = b2[0];
    // ---- scores: each thread handles 8 entities, k-loop coalesced over n ----
    for (int n = tid; n < N_ENT; n += 256) {
        float acc = 0.f;
        #pragma unroll 4
        for (int k = 0; k < KDIM; ++k) {
            float x = qs[k] + e_t[(size_t)k * N_ENT + n];
            acc = fmaf(gelu_exact(x), w2s[k], acc);
        }
        sc[n] = acc + bias2;
    }
    __syncthreads();

    // ---- top-32 via 32 rounds of LDS tree arg-max (ties -> lower index) ----
    for (int t = 0; t < MAX_K; ++t) {
        float best = -INFINITY;
        int   bi   = N_ENT;
        for (int n = tid; n < N_ENT; n += 256) {
            float v = sc[n];
            if (v > best) { best = v; bi = n; }
        }
        rmax[tid] = best;
        rarg[tid] = bi;
        __syncthreads();
        for (int s = 128; s > 0; s >>= 1) {
            if (tid < s) {
                float vo = rmax[tid + s], vm = rmax[tid];
                int   io = rarg[tid + s], im = rarg[tid];
                if (vo > vm || (vo == vm && io < im)) { rmax[tid] = vo; rarg[tid] = io; }
            }
            __syncthreads();
        }
        if (tid == 0) {
            int w = rarg[0];
            sidx[t] = w;
            out[OUT_SCORE_OFF + (size_t)bs * MAX_K + t] = rmax[0];
            out[OUT_MASK_OFF  + (size_t)bs * MAX_K + t] = 1.0f;
            sc[w] = -INFINITY;
        }
        __syncthreads();
    }

    // ---- gather chosen entity embeddings: [MAX_K, KDIM] per position ----
    for (int i = tid; i < MAX_K * KDIM; i += 256) {
        int t = i >> 7;
        int k = i & (KDIM - 1);
        out[OUT_EMB_OFF + ((size_t)bs * MAX_K + t) * KDIM + k] =
            ent[(size_t)sidx[t] * KDIM + k];
    }
}

// ---------------------------------------------------------------------------
// Launch
// ---------------------------------------------------------------------------
extern "C" void kernel_launch(void* const* d_in, const int* in_sizes, int n_in,
                              void* d_out, int out_size, void* d_ws, size_t ws_size,
                              hipStream_t stream) {
    const float* hs   = (const float*)d_in[0];   // [2,512,1024]
    const int*   eidx = (const int*)  d_in[1];   // [2048]
    const float* emb  = (const float*)d_in[2];   // [100000,128]
    const float* Wq   = (const float*)d_in[3];   // [1024,128]
    const float* bq   = (const float*)d_in[4];   // [128]
    const float* W1   = (const float*)d_in[5];   // [256,128]
    const float* b1   = (const float*)d_in[6];   // [128]
    const float* W2   = (const float*)d_in[7];   // [128,1]
    const float* b2   = (const float*)d_in[8];   // [1]
    float* out = (float*)d_out;

    // workspace layout (floats)
    float* Wc    = (float*)d_ws;                 // 1024*128
    float* bc    = Wc    + (size_t)H_DIM * KDIM; // 128
    float* qpart = bc    + KDIM;                 // 1024*128
    float* ent   = qpart + (size_t)BS * KDIM;    // 2048*128
    float* e_t   = ent   + (size_t)N_ENT * KDIM; // 128*2048

    // 1) gather entity embeddings
    gather_ent_kernel<<<(N_ENT * KDIM) / 256, 256, 0, stream>>>(emb, eidx, ent);

    // 2) fold projections: Wc = Wq @ W1a   (M=1024, N=128, K=128)
    wmma_gemm_f32_kernel<<<(64 * 8) / 4, 128, 0, stream>>>(
        Wq, W1, nullptr, Wc, H_DIM, KDIM, KDIM, 0);

    // 3) bc = bq @ W1a
    bias_fold_kernel<<<1, KDIM, 0, stream>>>(bq, W1, bc);

    // 4) q_part = HS @ Wc + bc             (M=1024, N=128, K=1024)
    wmma_gemm_f32_kernel<<<(64 * 8) / 4, 128, 0, stream>>>(
        hs, Wc, bc, qpart, BS, KDIM, H_DIM, 0);

    // 5) e_t = (ent @ W1b + b1)^T          (M=2048, N=128, K=128) stored [128,2048]
    wmma_gemm_f32_kernel<<<(128 * 8) / 4, 128, 0, stream>>>(
        ent, W1 + (size_t)KDIM * KDIM, b1, e_t, N_ENT, KDIM, KDIM, N_ENT);

    // 6) score + top-32 + gather, one block per (b,s)
    score_topk_kernel<<<BS, 256, 0, stream>>>(qpart, e_t, ent, W2, b2, out);
}